// Self_Attention_56066503082117
// MI455X (gfx1250) — compile-verified
//
#include <hip/hip_runtime.h>
#include <math.h>

typedef __attribute__((ext_vector_type(2))) float v2f;
typedef __attribute__((ext_vector_type(8))) float v8f;

constexpr int BATCH = 4;
constexpr int CH    = 64;    // channels
constexpr int CQ    = 8;     // C // 8 (q/k channels)
constexpr int NPIX  = 4096;  // H*W

// ---------------------------------------------------------------------------
// Kernel 1: fused 1x1-conv projections q = Wq x + bq, k = Wk x + bk, v = Wv x + bv
// One thread per (b, pixel). x channel loads are coalesced across threads.
// ---------------------------------------------------------------------------
__global__ void qkv_kernel(const float* __restrict__ x,
                           const float* __restrict__ Wq, const float* __restrict__ bq,
                           const float* __restrict__ Wk, const float* __restrict__ bk,
                           const float* __restrict__ Wv, const float* __restrict__ bv,
                           float* __restrict__ qb, float* __restrict__ kb,
                           float* __restrict__ vb) {
  int gid = blockIdx.x * blockDim.x + threadIdx.x;   // b*NPIX + n
  int b = gid >> 12;
  int n = gid & (NPIX - 1);
  const float* xb = x + ((size_t)b * CH) * NPIX + n;
  float xv[CH];
#pragma unroll
  for (int c = 0; c < CH; ++c) xv[c] = xb[(size_t)c * NPIX];

  for (int o = 0; o < CQ; ++o) {
    float aq = bq[o], ak = bk[o];
#pragma unroll
    for (int c = 0; c < CH; ++c) {
      aq = fmaf(Wq[o * CH + c], xv[c], aq);
      ak = fmaf(Wk[o * CH + c], xv[c], ak);
    }
    qb[((size_t)b * CQ + o) * NPIX + n] = aq;
    kb[((size_t)b * CQ + o) * NPIX + n] = ak;
  }
  for (int o = 0; o < CH; ++o) {
    float av = bv[o];
#pragma unroll
    for (int c = 0; c < CH; ++c) av = fmaf(Wv[o * CH + c], xv[c], av);
    vb[((size_t)b * CH + o) * NPIX + n] = av;
  }
}

// ---------------------------------------------------------------------------
// Kernel 2: S^T = K^T Q via V_WMMA_F32_16X16X4_F32 (K=8 -> 2 accumulating WMMAs).
// Storing S^T means rows are key-index j, contiguous in query-index i:
//   * D-fragment stores are coalesced 64B lines
//   * the buffer IS the (unnormalized) attention_map = A^T
// One wave handles a 16(j) x 64(i) strip: A-frag loaded once, 4 tiles.
// ---------------------------------------------------------------------------
__global__ void score_kernel(const float* __restrict__ qb,
                             const float* __restrict__ kb,
                             float* __restrict__ att) {
  const int lane = threadIdx.x & 31;
  const int lm = lane & 15;
  const int lh = lane >> 4;
  int wave = (blockIdx.x << 3) + (threadIdx.x >> 5);
  int b   = wave >> 14;          // 256*64 = 16384 waves per batch
  int rem = wave & 16383;
  int J   = rem >> 6;            // j-tile 0..255
  int Ist = rem & 63;            // i-strip, covers [Ist*64, Ist*64+64)

  const float* qB = qb + (size_t)b * CQ * NPIX;
  const float* kB = kb + (size_t)b * CQ * NPIX;
  float* attB = att + ((size_t)b * NPIX) * NPIX;

  // A-fragment: K-tile, A[m=j_local][kk=c]; a.x -> K=2*lh, a.y -> K=2*lh+1
  int jg = J * 16 + lm;
  v2f a0, a1;
  a0.x = kB[(size_t)(0 + 2 * lh) * NPIX + jg];
  a0.y = kB[(size_t)(1 + 2 * lh) * NPIX + jg];
  a1.x = kB[(size_t)(4 + 2 * lh) * NPIX + jg];
  a1.y = kB[(size_t)(5 + 2 * lh) * NPIX + jg];

#pragma unroll
  for (int t = 0; t < 4; ++t) {
    int ig = Ist * 64 + t * 16 + lm;
    v2f b0, b1;
    b0.x = qB[(size_t)(0 + 2 * lh) * NPIX + ig];
    b0.y = qB[(size_t)(1 + 2 * lh) * NPIX + ig];
    b1.x = qB[(size_t)(4 + 2 * lh) * NPIX + ig];
    b1.y = qB[(size_t)(5 + 2 * lh) * NPIX + ig];
    v8f acc = {};
    acc = __builtin_amdgcn_wmma_f32_16x16x4_f32(false, a0, false, b0,
                                                (short)0, acc, false, false);
    acc = __builtin_amdgcn_wmma_f32_16x16x4_f32(false, a1, false, b1,
                                                (short)0, acc, false, false);
#pragma unroll
    for (int r = 0; r < 8; ++r) {
      int j = J * 16 + r + 8 * lh;                     // D row
      attB[(size_t)j * NPIX + (Ist * 64 + t * 16 + lm)] = acc[r];
    }
  }
}

// ---------------------------------------------------------------------------
// Kernel 3: in-place softmax along each contiguous row of att (= softmax over
// query index of S). 16 values/thread held in registers -> 1 read + 1 write.
// ---------------------------------------------------------------------------
__global__ void softmax_kernel(float* __restrict__ att) {
  __shared__ float sred[256];
  const int t = threadIdx.x;
  float* p = att + (size_t)blockIdx.x * NPIX;

  float vals[16];
#pragma unroll
  for (int u = 0; u < 16; ++u) vals[u] = p[t + (u << 8)];

  float mx = vals[0];
#pragma unroll
  for (int u = 1; u < 16; ++u) mx = fmaxf(mx, vals[u]);
  sred[t] = mx; __syncthreads();
  for (int s = 128; s > 0; s >>= 1) {
    if (t < s) sred[t] = fmaxf(sred[t], sred[t + s]);
    __syncthreads();
  }
  mx = sred[0]; __syncthreads();

  float sum = 0.f;
#pragma unroll
  for (int u = 0; u < 16; ++u) { vals[u] = __expf(vals[u] - mx); sum += vals[u]; }
  sred[t] = sum; __syncthreads();
  for (int s = 128; s > 0; s >>= 1) {
    if (t < s) sred[t] += sred[t + s];
    __syncthreads();
  }
  float inv = 1.0f / sred[0];
#pragma unroll
  for (int u = 0; u < 16; ++u) p[t + (u << 8)] = vals[u] * inv;
}

// ---------------------------------------------------------------------------
// Kernel 4: o = V . A with A[n,m] = att[m][n], fused epilogue out = x + g*o.
// Block (8 waves) computes O[b][0:64][Mbase:Mbase+32]; each wave one 16x16
// tile, K-loop over n in steps of 4 (f32 WMMA), 16 WMMAs unrolled per chunk.
// attention rows are streamed (lane pairs read adjacent 16B) + prefetched.
// ---------------------------------------------------------------------------
__global__ void out_kernel(const float* __restrict__ vb,
                           const float* __restrict__ att,
                           const float* __restrict__ x,
                           const float* __restrict__ gamma,
                           float* __restrict__ out) {
  const int lane = threadIdx.x & 31;
  const int lm = lane & 15;
  const int lh = lane >> 4;
  const int w  = threadIdx.x >> 5;     // wave in block: 0..7
  const int ct = w & 3;                // c-tile 0..3
  const int mt = w >> 2;               // m-tile 0..1
  const int b  = blockIdx.x >> 7;      // 128 blocks per batch (N/32)
  const int Mbase = (blockIdx.x & 127) * 32 + mt * 16;

  // B-fragment source: lane lm -> output pixel row (Mbase+lm) of att
  const float* attRow = att + ((size_t)b * NPIX + Mbase + lm) * NPIX;
  // A-fragment source: lane lm -> channel row (ct*16+lm) of V
  const float* vRow = vb + ((size_t)b * CH + ct * 16 + lm) * NPIX;

  v8f acc = {};
  for (int k0 = 0; k0 < NPIX; k0 += 64) {
    __builtin_prefetch(attRow + k0 + 128, 0, 0);   // global_prefetch_b8
#pragma unroll
    for (int kk = 0; kk < 64; kk += 4) {
      int kb2 = k0 + kk + 2 * lh;
      v2f af, bf;
      af.x = vRow[kb2];
      af.y = vRow[kb2 + 1];
      bf.x = attRow[kb2];
      bf.y = attRow[kb2 + 1];
      acc = __builtin_amdgcn_wmma_f32_16x16x4_f32(false, af, false, bf,
                                                  (short)0, acc, false, false);
    }
  }

  float g = gamma[0];
#pragma unroll
  for (int r = 0; r < 8; ++r) {
    int c = ct * 16 + r + 8 * lh;      // D row -> channel
    int mg = Mbase + lm;               // D col -> pixel
    size_t idx = ((size_t)b * CH + c) * NPIX + mg;
    out[idx] = x[idx] + g * acc[r];    // coalesced 64B per row
  }
}

// ---------------------------------------------------------------------------
extern "C" void kernel_launch(void* const* d_in, const int* in_sizes, int n_in,
                              void* d_out, int out_size, void* d_ws, size_t ws_size,
                              hipStream_t stream) {
  const float* x     = (const float*)d_in[0];
  const float* Wq    = (const float*)d_in[1];
  const float* bq    = (const float*)d_in[2];
  const float* Wk    = (const float*)d_in[3];
  const float* bk    = (const float*)d_in[4];
  const float* Wv    = (const float*)d_in[5];
  const float* bv    = (const float*)d_in[6];
  const float* gamma = (const float*)d_in[7];

  float* out0 = (float*)d_out;                                // [B,C,H,W]
  float* att  = out0 + (size_t)BATCH * CH * NPIX;             // [B,N,N] = A^T

  float* qb = (float*)d_ws;                                   // [B,8,N]
  float* kb = qb + (size_t)BATCH * CQ * NPIX;                 // [B,8,N]
  float* vb = kb + (size_t)BATCH * CQ * NPIX;                 // [B,64,N]

  // 1) projections
  qkv_kernel<<<(BATCH * NPIX) / 256, 256, 0, stream>>>(x, Wq, bq, Wk, bk,
                                                       Wv, bv, qb, kb, vb);
  // 2) S^T tiles via f32 WMMA (8 waves/block, 16384 waves per batch)
  score_kernel<<<(BATCH * 256 * 64) / 8, 256, 0, stream>>>(qb, kb, att);
  // 3) row softmax in place -> att now holds attention_map = A^T
  softmax_kernel<<<BATCH * NPIX, 256, 0, stream>>>(att);
  // 4) o = V.A + residual epilogue via f32 WMMA
  out_kernel<<<BATCH * (NPIX / 32), 256, 0, stream>>>(vb, att, x, gamma, out0);
}